// MoebiusTransformer_57303453663743
// MI455X (gfx1250) — compile-verified
//
#include <hip/hip_runtime.h>
#include <math.h>

// MI455X roofline: 302 MB of traffic @ 23.3 TB/s => ~13 us floor; ~1 GFLOP f32
// compute is negligible. Pure streaming kernel: async b128 global<->LDS copies
// (ASYNCcnt path) for coalescing, wave32-friendly 3-float block consumption
// from LDS (bank-conflict free), deterministic per-row logdet reduction.

#define F_DIM   3072
#define NBLK    (F_DIM / 3)          // 1024 blocks of 3 per row
#define THREADS 256                  // 8 wave32 per workgroup
#define BPT     (NBLK / THREADS)     // 4 blocks per thread
#define ROWBYTES (F_DIM * 4)         // 12288 bytes per row buffer

__global__ __launch_bounds__(THREADS)
void moebius3_kernel(const float* __restrict__ x,
                     const float* __restrict__ w,
                     float* __restrict__ y,
                     float* __restrict__ logdet)
{
    __shared__ float smem[2 * F_DIM + THREADS];  // [x|y row][w row][reduction]

    const int row = blockIdx.x;
    const int t   = threadIdx.x;

    // LDS byte offsets: for generic pointers derived from an LDS object, the
    // low 32 bits are the wave-relative LDS offset (addrspacecast local->flat
    // places the offset in the low dword).
    const unsigned lds_base = (unsigned)(unsigned long long)(void*)&smem[0];
    const unsigned voff     = (unsigned)t * 16u;          // per-lane byte chunk
    const unsigned ldsx     = lds_base + voff;            // x/y buffer
    const unsigned ldsw     = lds_base + ROWBYTES + voff; // w buffer

    const unsigned long long xrow = (unsigned long long)(x + (size_t)row * F_DIM);
    const unsigned long long wrow = (unsigned long long)(w + (size_t)row * F_DIM);
    const unsigned long long yrow = (unsigned long long)(y + (size_t)row * F_DIM);

    // Async global->LDS: 3 x-chunks + 3 w-chunks of b128 per thread.
    // ISA: inst offset is added to BOTH the LDS addr (VDST vgpr) and the
    // global addr (SADDR + VADDR), so chunk k lands at +k*4096 on both sides.
    asm volatile(
        "global_load_async_to_lds_b128 %0, %1, %2 offset:0\n\t"
        "global_load_async_to_lds_b128 %0, %1, %2 offset:4096\n\t"
        "global_load_async_to_lds_b128 %0, %1, %2 offset:8192"
        :: "v"(ldsx), "v"(voff), "s"(xrow) : "memory");
    asm volatile(
        "global_load_async_to_lds_b128 %0, %1, %2 offset:0\n\t"
        "global_load_async_to_lds_b128 %0, %1, %2 offset:4096\n\t"
        "global_load_async_to_lds_b128 %0, %1, %2 offset:8192"
        :: "v"(ldsw), "v"(voff), "s"(wrow) : "memory");

    asm volatile("s_wait_asynccnt 0" ::: "memory");
    __syncthreads();

    float* sx = smem;           // becomes y in-place
    float* sw = smem + F_DIM;

    float acc = 0.0f;
#pragma unroll
    for (int k = 0; k < BPT; ++k) {
        const int j = t + k * THREADS;   // block index: dword 3j -> bank 3t mod 64
        const int o = 3 * j;

        float x0 = sx[o], x1 = sx[o + 1], x2 = sx[o + 2];
        float w0 = sw[o], w1 = sw[o + 1], w2 = sw[o + 2];

        float r2    = x0 * x0 + x1 * x1 + x2 * x2;
        float r     = sqrtf(r2);
        float inv_r = 1.0f / r;

        float xn0 = x0 * inv_r, xn1 = x1 * inv_r, xn2 = x2 * inv_r;
        float p0 = xn0 + w0, p1 = xn1 + w1, p2 = xn2 + w2;   // xn + w

        float wns = w0 * w0 + w1 * w1 + w2 * w2;             // |w|^2
        float pns = p0 * p0 + p1 * p1 + p2 * p2;             // |xn+w|^2
        float inv_pns = 1.0f / pns;
        float c   = (1.0f - wns) * inv_pns;

        float yn0 = c * p0 + w0, yn1 = c * p1 + w1, yn2 = c * p2 + w2;

        // y = r * yn, written in place over x (thread-private block)
        sx[o]     = r * yn0;
        sx[o + 1] = r * yn1;
        sx[o + 2] = r * yn2;

        // Jacobian J = c*I + u*xn^T - a*p*q^T
        //   q = p - (xn.p)*xn,  a = 2c/pns,  u = yn - c*xn
        float s  = xn0 * p0 + xn1 * p1 + xn2 * p2;
        float q0 = p0 - s * xn0, q1 = p1 - s * xn1, q2 = p2 - s * xn2;
        float a  = 2.0f * c * inv_pns;
        float u0 = yn0 - c * xn0, u1 = yn1 - c * xn1, u2 = yn2 - c * xn2;

        // det(c*I + U V^T) = c * det(c*I2 + V^T U), U=[u|-a p], V=[xn|q]
        float xu = xn0 * u0 + xn1 * u1 + xn2 * u2;   // xn.u
        float qp = q0 * p0 + q1 * p1 + q2 * p2;      // q.p
        float qu = q0 * u0 + q1 * u1 + q2 * u2;      // q.u
        float det = c * ((c + xu) * (c - a * qp) + a * s * qu);

        acc += logf(fabsf(det));
    }

    __syncthreads();   // all y blocks written to LDS

    // Async LDS->global store of the y row, b128 coalesced
    asm volatile(
        "global_store_async_from_lds_b128 %0, %1, %2 offset:0\n\t"
        "global_store_async_from_lds_b128 %0, %1, %2 offset:4096\n\t"
        "global_store_async_from_lds_b128 %0, %1, %2 offset:8192"
        :: "v"(voff), "v"(ldsx), "s"(yrow) : "memory");

    // Deterministic per-row log|det| reduction (no fp atomics)
    float* red = smem + 2 * F_DIM;
    red[t] = acc;
    __syncthreads();
#pragma unroll
    for (int off = THREADS / 2; off > 0; off >>= 1) {
        if (t < off) red[t] += red[t + off];
        __syncthreads();
    }
    if (t == 0) logdet[row] = red[0];

    asm volatile("s_wait_asynccnt 0" ::: "memory");
}

extern "C" void kernel_launch(void* const* d_in, const int* in_sizes, int n_in,
                              void* d_out, int out_size, void* d_ws, size_t ws_size,
                              hipStream_t stream)
{
    const float* x = (const float*)d_in[0];
    const float* w = (const float*)d_in[1];
    float* out = (float*)d_out;

    const int total = in_sizes[0];        // B * F
    const int nrows = out_size - total;   // B  (out = y flat ++ logdet)
    // F is compiled in (F_DIM = 3072); reference uses B=8192, F=3072.

    float* yout = out;
    float* ld   = out + (size_t)total;

    moebius3_kernel<<<nrows, THREADS, 0, stream>>>(x, w, yout, ld);
}